// IMUFwd_14053132993160
// MI455X (gfx1250) — compile-verified
//
#include <hip/hip_runtime.h>
#include <math.h>

// Problem constants (fixed by the reference).
#define N_SAMPLES 262144
#define M_POSES   2048
#define NCHUNK    256            // number of chunks == grid size of passes 1/3/5
#define TPB       256            // threads per block (8 wave32s)
#define EPT       4              // elements per thread
#define CHUNK     (TPB * EPT)    // 1024 elements per chunk
#define GRAVITY   9.81007f

typedef __attribute__((ext_vector_type(2))) float v2f;
typedef __attribute__((ext_vector_type(8))) float v8f;

// ---------------- quaternion helpers (q = (x,y,z,w)) ----------------

__device__ __forceinline__ float4 qmulq(float4 a, float4 b) {
    return make_float4(
        a.w * b.x + a.x * b.w + a.y * b.z - a.z * b.y,
        a.w * b.y - a.x * b.z + a.y * b.w + a.z * b.x,
        a.w * b.z + a.x * b.y - a.y * b.x + a.z * b.w,
        a.w * b.w - a.x * b.x - a.y * b.y - a.z * b.z);
}

__device__ __forceinline__ float4 qnormalize(float4 q) {
    float n = rsqrtf(q.x * q.x + q.y * q.y + q.z * q.z + q.w * q.w);
    return make_float4(q.x * n, q.y * n, q.z * n, q.w * n);
}

// exp_so3(phi): xyz = phi * sin(|phi|/2)/|phi|, w = cos(|phi|/2)
__device__ __forceinline__ float4 expso3(float px, float py, float pz) {
    float t2 = px * px + py * py + pz * pz;
    float th = sqrtf(t2);
    float h  = 0.5f * th;
    float s  = (th < 1e-6f) ? (0.5f - t2 * (1.0f / 48.0f)) : (sinf(h) / th);
    return make_float4(px * s, py * s, pz * s, cosf(h));
}

// v' = v + 2w(xyz x v) + 2 xyz x (xyz x v)
__device__ __forceinline__ void qrotv(float4 q, float vx, float vy, float vz,
                                      float& ox, float& oy, float& oz) {
    float tx = 2.f * (q.y * vz - q.z * vy);
    float ty = 2.f * (q.z * vx - q.x * vz);
    float tz = 2.f * (q.x * vy - q.y * vx);
    ox = vx + q.w * tx + (q.y * tz - q.z * ty);
    oy = vy + q.w * ty + (q.z * tx - q.x * tz);
    oz = vz + q.w * tz + (q.x * ty - q.y * tx);
}

// Left-multiplication matrix L(q): L(q) * r(col-vec x,y,z,w) == q (x) r.
// Row-major 4x4 in 16 floats.
__device__ __forceinline__ void writeL(float* Mm, float4 q) {
    Mm[0]  =  q.w; Mm[1]  = -q.z; Mm[2]  =  q.y; Mm[3]  =  q.x;
    Mm[4]  =  q.z; Mm[5]  =  q.w; Mm[6]  = -q.x; Mm[7]  =  q.y;
    Mm[8]  = -q.y; Mm[9]  =  q.x; Mm[10] =  q.w; Mm[11] =  q.z;
    Mm[12] = -q.x; Mm[13] = -q.y; Mm[14] = -q.z; Mm[15] =  q.w;
}

__device__ __forceinline__ void matmul4(float* D, const float* L, const float* R) {
    #pragma unroll
    for (int r = 0; r < 4; ++r)
        #pragma unroll
        for (int c = 0; c < 4; ++c) {
            float s = 0.f;
            #pragma unroll
            for (int k = 0; k < 4; ++k) s += L[r * 4 + k] * R[k * 4 + c];
            D[r * 4 + c] = s;
        }
}

// ======================= Pass 1: per-chunk quaternion block scan =======================
// Each thread multiplies its 4 local delta-quats, then a block Hillis-Steele scan
// produces thread-exclusive prefixes within the chunk and the chunk aggregate.
__global__ __launch_bounds__(TPB) void k_pass1(const float* __restrict__ gyr,
                                               const float* __restrict__ gb,
                                               const float* __restrict__ dts,
                                               float4* __restrict__ thrQ,
                                               float4* __restrict__ chunkQ) {
    __shared__ float4 sq[TPB];
    const int t  = threadIdx.x;
    const int gt = blockIdx.x * TPB + t;
    const float gbx = gb[0], gby = gb[1], gbz = gb[2];
    float4 q = make_float4(0.f, 0.f, 0.f, 1.f);
    const int base = gt * EPT;
    #pragma unroll
    for (int e = 0; e < EPT; ++e) {
        int i = base + e;
        float dt = dts[i];
        float4 dq = expso3((gyr[3 * i] - gbx) * dt,
                           (gyr[3 * i + 1] - gby) * dt,
                           (gyr[3 * i + 2] - gbz) * dt);
        q = qmulq(q, dq);
    }
    sq[t] = q;
    __syncthreads();
    for (int off = 1; off < TPB; off <<= 1) {
        float4 prev;
        bool has = (t >= off);
        if (has) prev = sq[t - off];
        __syncthreads();
        if (has) sq[t] = qmulq(prev, sq[t]);   // earlier on the left
        __syncthreads();
    }
    thrQ[gt] = (t == 0) ? make_float4(0.f, 0.f, 0.f, 1.f) : sq[t - 1];
    if (t == TPB - 1) chunkQ[blockIdx.x] = qnormalize(sq[t]);
}

// ======================= Pass 2: scan of 256 chunk quaternions via WMMA =======================
// Quaternion products as 4x4 L-matrix products; Hillis-Steele scan with double-buffered
// LDS. Each V_WMMA_F32_16X16X4_F32 computes 4 independent 4x4 products:
//   A (16x4)  = [L0;L1;L2;L3] stacked rows, B (4x16) = [R0 R1 R2 R3] side by side,
//   diagonal 4x4 blocks of D (16x16) = Li*Ri.
// f32 A 16x4 layout: lanes 0-15 rows, vgpr0/1 = K{0,1} (lanes<16) / K{2,3} (lanes>=16).
// f32 B 4x16 assumed mirrored: lanes = cols, vgpr0/1 = K{0,1} / K{2,3} split by half-wave.
// f32 D 16x16 layout: lane = col N (mod 16), vgpr r -> row r (lanes<16) / r+8 (lanes>=16).
__global__ __launch_bounds__(32) void k_pass2(const float4* __restrict__ chunkQ,
                                              const float* __restrict__ q0p,
                                              float4* __restrict__ qin) {
    __shared__ float buf[2][NCHUNK * 16];   // 2 x 16 KiB double buffer
    const int lane = threadIdx.x;
    for (int c = lane; c < NCHUNK; c += 32) {
        float4 q = (c == 0) ? make_float4(q0p[0], q0p[1], q0p[2], q0p[3]) : chunkQ[c - 1];
        writeL(&buf[0][c * 16], q);
    }
    __syncthreads();
    int cur = 0;
    const int half = lane >> 4, m = lane & 15, i4 = m >> 2, rr = m & 3, kb = half * 2;
    for (int s = 1; s < NCHUNK; s <<= 1) {
        int dst = cur ^ 1;
        // indices < s carry over unchanged
        for (int j = lane; j < s * 16; j += 32) buf[dst][j] = buf[cur][j];
        // remainder (so the WMMA loop always handles full groups of 4)
        int rem = (NCHUNK - s) & 3;
        if (lane == 0) {
            for (int idx = s; idx < s + rem; ++idx)
                matmul4(&buf[dst][idx * 16], &buf[cur][(idx - s) * 16], &buf[cur][idx * 16]);
        }
        int first   = s + rem;
        int ngroups = (NCHUNK - first) >> 2;
        for (int g = 0; g < ngroups; ++g) {
            int bidx = first + 4 * g;                       // dst[bidx+i] = src[bidx+i-s]*src[bidx+i]
            const float* Lm = &buf[cur][(bidx + i4 - s) * 16];
            const float* Rm = &buf[cur][(bidx + i4) * 16];
            v2f a, b;
            a[0] = Lm[rr * 4 + kb];       a[1] = Lm[rr * 4 + kb + 1];
            b[0] = Rm[kb * 4 + rr];       b[1] = Rm[(kb + 1) * 4 + rr];
            v8f cacc = {};
            v8f d = __builtin_amdgcn_wmma_f32_16x16x4_f32(
                false, a, false, b, (short)0, cacc, false, false);
            // extract diagonal 4x4 block (i4,i4), column rr, rows 4*i4..4*i4+3
            float* Dm = &buf[dst][(bidx + i4) * 16 + rr];
            if (half == 0) {
                if (m < 4)        { Dm[0] = d[0]; Dm[4] = d[1]; Dm[8]  = d[2]; Dm[12] = d[3]; }
                else if (m < 8)   { Dm[0] = d[4]; Dm[4] = d[5]; Dm[8]  = d[6]; Dm[12] = d[7]; }
            } else {
                if (m >= 12)      { Dm[0] = d[4]; Dm[4] = d[5]; Dm[8]  = d[6]; Dm[12] = d[7]; }
                else if (m >= 8)  { Dm[0] = d[0]; Dm[4] = d[1]; Dm[8]  = d[2]; Dm[12] = d[3]; }
            }
        }
        __syncthreads();
        cur = dst;
    }
    // q = L * (0,0,0,1)^T = column 3
    for (int c = lane; c < NCHUNK; c += 32) {
        const float* Mm = &buf[cur][c * 16];
        qin[c] = qnormalize(make_float4(Mm[3], Mm[7], Mm[11], Mm[15]));
    }
}

// ======================= Pass 3: per-element rotation replay + (A,T,P) block scan ===========
__global__ __launch_bounds__(TPB) void k_pass3(const float* __restrict__ acc,
                                               const float* __restrict__ gyr,
                                               const float* __restrict__ ab,
                                               const float* __restrict__ gb,
                                               const float* __restrict__ dts,
                                               const float4* __restrict__ qin,
                                               const float4* __restrict__ thrQ,
                                               float4* __restrict__ rots,
                                               float4* __restrict__ aw,
                                               float4* __restrict__ thrATP,
                                               float4* __restrict__ chunkATP) {
    __shared__ float4 sA[TPB], sP[TPB];
    const int t = threadIdx.x, ch = blockIdx.x;
    const int gt = ch * TPB + t;
    const float abx = ab[0], aby = ab[1], abz = ab[2];
    const float gbx = gb[0], gby = gb[1], gbz = gb[2];
    float4 q = qnormalize(qmulq(qin[ch], thrQ[gt]));   // q_{i-1} at segment start
    float Ax = 0, Ay = 0, Az = 0, Tt = 0, Px = 0, Py = 0, Pz = 0;
    const int base = gt * EPT;
    #pragma unroll
    for (int e = 0; e < EPT; ++e) {
        int i = base + e;
        float dt = dts[i];
        float ax = acc[3 * i] - abx, ay = acc[3 * i + 1] - aby, az = acc[3 * i + 2] - abz;
        float wx, wy, wz;
        qrotv(q, ax, ay, az, wx, wy, wz);
        wz -= GRAVITY;                                  // a_world
        aw[i] = make_float4(wx, wy, wz, 0.f);
        float hdt2 = 0.5f * dt * dt;
        Px += Ax * dt + wx * hdt2;  Py += Ay * dt + wy * hdt2;  Pz += Az * dt + wz * hdt2;
        Ax += wx * dt;              Ay += wy * dt;              Az += wz * dt;
        Tt += dt;
        float4 dq = expso3((gyr[3 * i] - gbx) * dt,
                           (gyr[3 * i + 1] - gby) * dt,
                           (gyr[3 * i + 2] - gbz) * dt);
        q = qnormalize(qmulq(q, dq));
        rots[i] = q;
    }
    sA[t] = make_float4(Ax, Ay, Az, Tt);
    sP[t] = make_float4(Px, Py, Pz, 0.f);
    __syncthreads();
    for (int off = 1; off < TPB; off <<= 1) {
        float4 pa, pp;
        bool has = (t >= off);
        if (has) { pa = sA[t - off]; pp = sP[t - off]; }
        __syncthreads();
        if (has) {
            float4 ca = sA[t], cp = sP[t];
            sA[t] = make_float4(pa.x + ca.x, pa.y + ca.y, pa.z + ca.z, pa.w + ca.w);
            sP[t] = make_float4(pp.x + pa.x * ca.w + cp.x,
                                pp.y + pa.y * ca.w + cp.y,
                                pp.z + pa.z * ca.w + cp.z, 0.f);
        }
        __syncthreads();
    }
    float4 exA = (t == 0) ? make_float4(0, 0, 0, 0) : sA[t - 1];
    float4 exP = (t == 0) ? make_float4(0, 0, 0, 0) : sP[t - 1];
    thrATP[2 * gt] = exA;
    thrATP[2 * gt + 1] = exP;
    if (t == TPB - 1) { chunkATP[2 * ch] = sA[t]; chunkATP[2 * ch + 1] = sP[t]; }
}

// ======================= Pass 4: scan of 256 chunk affine (v,p) maps =======================
__global__ void k_pass4(const float4* __restrict__ chunkATP,
                        const float* __restrict__ v0p, const float* __restrict__ p0p,
                        float4* __restrict__ chunkVP) {
    if (threadIdx.x != 0 || blockIdx.x != 0) return;
    float vx = v0p[0], vy = v0p[1], vz = v0p[2];
    float px = p0p[0], py = p0p[1], pz = p0p[2];
    for (int c = 0; c < NCHUNK; ++c) {
        chunkVP[2 * c]     = make_float4(vx, vy, vz, 0.f);
        chunkVP[2 * c + 1] = make_float4(px, py, pz, 0.f);
        float4 A = chunkATP[2 * c], P = chunkATP[2 * c + 1];
        px += vx * A.w + P.x;  py += vy * A.w + P.y;  pz += vz * A.w + P.z;
        vx += A.x;             vy += A.y;             vz += A.z;
    }
}

// ======================= Pass 5: per-element (v,p) replay =======================
__global__ __launch_bounds__(TPB) void k_pass5(const float* __restrict__ dts,
                                               const float4* __restrict__ aw,
                                               const float4* __restrict__ thrATP,
                                               const float4* __restrict__ chunkVP,
                                               float4* __restrict__ poss) {
    const int t = threadIdx.x, ch = blockIdx.x;
    const int gt = ch * TPB + t;
    float4 vin = chunkVP[2 * ch], pin = chunkVP[2 * ch + 1];
    float4 exA = thrATP[2 * gt], exP = thrATP[2 * gt + 1];
    float vx = vin.x + exA.x, vy = vin.y + exA.y, vz = vin.z + exA.z;
    float px = pin.x + vin.x * exA.w + exP.x;
    float py = pin.y + vin.y * exA.w + exP.y;
    float pz = pin.z + vin.z * exA.w + exP.z;
    const int base = gt * EPT;
    #pragma unroll
    for (int e = 0; e < EPT; ++e) {
        int i = base + e;
        float dt = dts[i];
        float4 a = aw[i];
        float hdt2 = 0.5f * dt * dt;
        px += vx * dt + a.x * hdt2;  py += vy * dt + a.y * hdt2;  pz += vz * dt + a.z * hdt2;
        poss[i] = make_float4(px, py, pz, 0.f);
        vx += a.x * dt;  vy += a.y * dt;  vz += a.z * dt;
    }
}

// ======================= Pass 6: gather at sync + error reduction =======================
__global__ __launch_bounds__(TPB) void k_pass6(const float4* __restrict__ rots,
                                               const float4* __restrict__ poss,
                                               const float* __restrict__ prot,
                                               const float* __restrict__ ptrans,
                                               const int* __restrict__ sync,
                                               float* __restrict__ out) {
    __shared__ float sl[TPB], st2[TPB];
    const int t = threadIdx.x;
    float l2 = 0.f, t2 = 0.f;
    for (int mI = t; mI < M_POSES; mI += TPB) {
        int s = sync[mI];
        float4 r = rots[s], p = poss[s];
        float4 pr = make_float4(-prot[4 * mI], -prot[4 * mI + 1], -prot[4 * mI + 2], prot[4 * mI + 3]);
        float4 rel = qmulq(pr, r);
        float nn = sqrtf(rel.x * rel.x + rel.y * rel.y + rel.z * rel.z);
        float th = 2.f * atan2f(nn, rel.w);
        float sc = (nn < 1e-7f) ? 2.f : th / fmaxf(nn, 1e-12f);
        float lx = rel.x * sc, ly = rel.y * sc, lz = rel.z * sc;
        l2 += lx * lx + ly * ly + lz * lz;
        float dx = ptrans[3 * mI] - p.x, dy = ptrans[3 * mI + 1] - p.y, dz = ptrans[3 * mI + 2] - p.z;
        t2 += dx * dx + dy * dy + dz * dz;
    }
    sl[t] = l2; st2[t] = t2;
    __syncthreads();
    for (int off = TPB >> 1; off > 0; off >>= 1) {
        if (t < off) { sl[t] += sl[t + off]; st2[t] += st2[t + off]; }
        __syncthreads();
    }
    if (t == 0) out[0] = sqrtf(sl[0]) + st2[0] / (float)(M_POSES * 3);
}

// ======================= host glue =======================
extern "C" void kernel_launch(void* const* d_in, const int* in_sizes, int n_in,
                              void* d_out, int out_size, void* d_ws, size_t ws_size,
                              hipStream_t stream) {
    const float* accels      = (const float*)d_in[0];
    const float* gyros       = (const float*)d_in[1];
    const float* accel_bias  = (const float*)d_in[2];
    const float* gyro_bias   = (const float*)d_in[3];
    const float* dts         = (const float*)d_in[4];
    const float* init_rot    = (const float*)d_in[5];
    const float* init_pos    = (const float*)d_in[6];
    const float* init_vel    = (const float*)d_in[7];
    const float* poses_rot   = (const float*)d_in[8];
    const float* poses_trans = (const float*)d_in[9];
    const int*   sync        = (const int*)d_in[10];
    float* out = (float*)d_out;

    // Workspace layout (~15.1 MiB total)
    const int NTHR = NCHUNK * TPB;            // 65536 threads in passes 1/3/5
    char* w = (char*)d_ws;
    float4* thrQ     = (float4*)w;                       w += (size_t)NTHR * 16;      // 1 MiB
    float4* chunkQ   = (float4*)w;                       w += (size_t)NCHUNK * 16;
    float4* qin      = (float4*)w;                       w += (size_t)NCHUNK * 16;
    float4* rots     = (float4*)w;                       w += (size_t)N_SAMPLES * 16; // 4 MiB
    float4* aw       = (float4*)w;                       w += (size_t)N_SAMPLES * 16; // 4 MiB
    float4* thrATP   = (float4*)w;                       w += (size_t)NTHR * 32;      // 2 MiB
    float4* chunkATP = (float4*)w;                       w += (size_t)NCHUNK * 32;
    float4* chunkVP  = (float4*)w;                       w += (size_t)NCHUNK * 32;
    float4* poss     = (float4*)w;                       w += (size_t)N_SAMPLES * 16; // 4 MiB
    (void)ws_size; (void)in_sizes; (void)n_in; (void)out_size;

    k_pass1<<<NCHUNK, TPB, 0, stream>>>(gyros, gyro_bias, dts, thrQ, chunkQ);
    k_pass2<<<1, 32, 0, stream>>>(chunkQ, init_rot, qin);
    k_pass3<<<NCHUNK, TPB, 0, stream>>>(accels, gyros, accel_bias, gyro_bias, dts,
                                        qin, thrQ, rots, aw, thrATP, chunkATP);
    k_pass4<<<1, 32, 0, stream>>>(chunkATP, init_vel, init_pos, chunkVP);
    k_pass5<<<NCHUNK, TPB, 0, stream>>>(dts, aw, thrATP, chunkVP, poss);
    k_pass6<<<1, TPB, 0, stream>>>(rots, poss, poses_rot, poses_trans, sync, out);
}